// _FinDiffNonUniform_51608327029442
// MI455X (gfx1250) — compile-verified
//
#include <hip/hip_runtime.h>
#include <hip/hip_bf16.h>
#include <stdint.h>

// FinDiffNonUniform: yd[i,b] = sum_s coef[i,s] * y[i+off[i,s], b]
// Banded GEMM per 16-row tile: D(16x16) = A(16x32) x Y(32x16) via 8 chained
// V_WMMA_F32_16X16X4_F32. Y window staged straight into LDS with CDNA5
// async-to-LDS loads (ASYNCcnt); A (banded coeffs) built once per block.

constexpr int kN = 8192;
constexpr int kB = 4096;
constexpr int kS = 7;

typedef float v2f __attribute__((ext_vector_type(2)));
typedef float v8f __attribute__((ext_vector_type(8)));

constexpr int YSTRIDE = 80;  // 32-row y window, padded row stride (floats)
constexpr int ASTRIDE = 34;  // 16x32 banded coeff matrix, padded stride (floats)

__global__ __launch_bounds__(128)
void _FinDiffNonUniform_51608327029442_kernel(const float* __restrict__ y,
                                              const float* __restrict__ coef,
                                              const int*   __restrict__ offs,
                                              float*       __restrict__ out) {
  __shared__ float Ylds[32 * YSTRIDE];
  __shared__ float Alds[16 * ASTRIDE];

  const int tid  = threadIdx.x;
  const int wave = tid >> 5;
  const int lane = tid & 31;
  const int half = lane >> 4;   // 0: lanes 0-15, 1: lanes 16-31
  const int n    = lane & 15;

  const int t        = blockIdx.y * 16;     // first output row of tile
  const int colBlock = blockIdx.x * 64;     // first column of block

  int base = t - 3;
  if (base < 0) base = 0;
  if (base > kN - 32) base = kN - 32;       // all i+off land in [base, base+32)

  // ---- stage Y[base..base+32) x 64 cols directly into LDS (async, 16B/lane) ----
  #pragma unroll
  for (int it = 0; it < 4; ++it) {
    int e   = it * 128 + tid;               // 512 float4 elements total
    int row = e >> 4;                       // 16 float4 per row
    int c4  = e & 15;
    const float* gp = y + (size_t)(base + row) * kB + colBlock + c4 * 4;
    unsigned lds_off =
        (unsigned)reinterpret_cast<uintptr_t>(&Ylds[row * YSTRIDE + c4 * 4]);
    asm volatile("global_load_async_to_lds_b128 %0, %1, off"
                 :: "v"(lds_off), "v"((unsigned long long)(uintptr_t)gp)
                 : "memory");
  }

  // ---- zero banded-coefficient matrix A ----
  #pragma unroll
  for (int it = 0; it < 5; ++it) {
    int e = it * 128 + tid;
    if (e < 16 * ASTRIDE) Alds[e] = 0.0f;
  }

  // wait for this wave's async fills, then sync the workgroup
  asm volatile("s_wait_asynccnt 0x0" ::: "memory");
  __syncthreads();

  // ---- scatter the 7 taps of each of the 16 rows into A ----
  if (tid < 16 * kS) {
    int m = tid / kS;
    int s = tid - m * kS;
    int i = t + m;
    int off = offs[i * kS + s];
    float c = coef[i * kS + s];
    int k = i + off - base;                 // guaranteed in [0, 32)
    Alds[m * ASTRIDE + k] = c;
  }
  __syncthreads();

  // ---- per wave: 16x16 tile, D = A(16x32) * Y(32x16) via 8 f32 WMMAs ----
  v8f acc = {0.f, 0.f, 0.f, 0.f, 0.f, 0.f, 0.f, 0.f};
  const int col0 = colBlock + wave * 16;
  const int ncol = wave * 16 + n;           // column within staged window

  #pragma unroll
  for (int kk = 0; kk < 8; ++kk) {
    int kb = kk * 4 + half * 2;             // this lane's K pair: kb, kb+1
    v2f a, b;
    // A 16x4 fragment: lanes 0-15 -> M=lane, K={4kk,4kk+1}; lanes 16-31 -> K={4kk+2,4kk+3}
    a.x = Alds[n * ASTRIDE + kb];
    a.y = Alds[n * ASTRIDE + kb + 1];
    // B 4x16 fragment: mirrored row striping, N = lane%16
    b.x = Ylds[kb * YSTRIDE + ncol];
    b.y = Ylds[(kb + 1) * YSTRIDE + ncol];
    acc = __builtin_amdgcn_wmma_f32_16x16x4_f32(
        false, a, false, b, (short)0, acc, false, false);
  }

  // ---- store C tile: VGPR v -> row (v + 8*half), col = lane%16 ----
  #pragma unroll
  for (int v = 0; v < 8; ++v) {
    int row = t + v + half * 8;
    out[(size_t)row * kB + col0 + n] = acc[v];
  }
}

extern "C" void kernel_launch(void* const* d_in, const int* in_sizes, int n_in,
                              void* d_out, int out_size, void* d_ws, size_t ws_size,
                              hipStream_t stream) {
  (void)in_sizes; (void)n_in; (void)out_size; (void)d_ws; (void)ws_size;
  const float* y    = (const float*)d_in[0];
  const float* coef = (const float*)d_in[1];
  const int*   offs = (const int*)d_in[2];
  float* out = (float*)d_out;

  dim3 block(128);
  dim3 grid(kB / 64, kN / 16);
  _FinDiffNonUniform_51608327029442_kernel<<<grid, block, 0, stream>>>(y, coef, offs, out);
}